// BConv2d_73581379715141
// MI455X (gfx1250) — compile-verified
//
#include <hip/hip_runtime.h>

// ---------------------------------------------------------------------------
// Butterfly conv (B=64, IC=16, OC=32, N=1024, 10 stages) for MI455X (gfx1250).
//
// Strategy (see analysis): per workgroup = (oc, 16-batch tile), keep the whole
// 1024-point line in LDS, compose stages {0..3} and {4..7} into dense 16x16
// matrices built on the fly, and apply them with chains of
// v_wmma_f32_16x16x4_f32 (K=16 -> 4 chained WMMAs). Stages {8,9} are a 4x4
// mix done with VALU. Accumulate the ic-mean in LDS, add bias, store.
// ---------------------------------------------------------------------------

typedef __attribute__((ext_vector_type(2))) float v2f;
typedef __attribute__((ext_vector_type(4))) float f4;
typedef __attribute__((ext_vector_type(8))) float v8f;

#define NPTS   1024
#define YSTR   17            // padded row stride (floats) -> conflict-free LDS
#define Y_OFF  0
#define A_OFF  (NPTS * YSTR)           // 17408
#define M_OFF  (2 * NPTS * YSTR)       // 34816
#define MSLOT  (16 * YSTR)             // 272 floats per per-wave matrix slot
#define SHM_FLOATS (2 * NPTS * YSTR + 8 * MSLOT)   // 36992 floats = 147968 B

__global__ __launch_bounds__(256) void bconv_butterfly_kernel(
    const float* __restrict__ x,        // [16][64][1024] flat
    const float* __restrict__ tw,       // [16][32][10][512][2][2] flat
    const float* __restrict__ bias,     // [32][1024] flat
    float* __restrict__ out)            // [32][64][1024] flat
{
    extern __shared__ float smem[];
    float* Y    = smem + Y_OFF;   // Y[n][b], stride YSTR
    float* ACC  = smem + A_OFF;   // ACC[n][b], stride YSTR
    float* MATS = smem + M_OFF;   // 8 slots of 16x16 (stride YSTR rows)

    const int t     = threadIdx.x;
    const int lane  = t & 31;
    const int wv    = t >> 5;          // wave id 0..7
    const int mlane = lane & 15;       // M (A) / N (B,D) index
    const int kh    = lane >> 4;       // lane-half -> K offset 2*kh, row +8*kh
    const int b0    = blockIdx.x * 16; // batch tile base
    const int o     = blockIdx.y;      // output channel

    // zero the ic-accumulator
    for (int idx = t; idx < 16384; idx += 256) {
        int n = idx & 1023, b = idx >> 10;
        ACC[n * YSTR + b] = 0.f;
    }

    float* mymat = MATS + wv * MSLOT;

    for (int i = 0; i < 16; ++i) {
        __syncthreads();
        const float* xi   = x + i * 65536;
        const float* twio = tw + (size_t)(i * 32 + o) * (10 * 512 * 4);

        // prefetch this (ic,oc) twiddle slice (80 KB) -> global_prefetch_b8
        for (int off = t * 128; off < 10 * 512 * 4 * 4; off += 256 * 128)
            __builtin_prefetch((const char*)twio + off, 0, 1);

        // load x (coalesced over m) and scatter bit-reversed into LDS:
        // Y[n][b] = x[i][b0+b][rev10(n)]  (rev10 is an involution)
        for (int idx = t; idx < 16384; idx += 256) {
            int m = idx & 1023;
            int b = idx >> 10;
            float v = xi[(b0 + b) * 1024 + m];
            int n = __brev((unsigned)m) >> 22;   // 10-bit reversal
            Y[n * YSTR + b] = v;
        }
        __syncthreads();

        // ---------------- Phase A: stages 0..3 (mix n[3:0]) ----------------
        // One dense 16x16 matrix per k1 = n[9:4]; wave wv handles k1 = blk*8+wv.
        for (int blk = 0; blk < 8; ++blk) {
            int k1 = blk * 8 + wv;
            // build combined matrix = S3*S2*S1*S0 applied to identity
            for (int e = lane; e < 256; e += 32) {
                int p = e >> 4, q = e & 15;
                mymat[p * YSTR + q] = (p == q) ? 1.f : 0.f;
            }
            for (int l = 0; l < 4; ++l) {
                const float* twl = twio + l * 2048;        // stage l, 512*(2x2)
                for (int tt = lane; tt < 128; tt += 32) {  // 8 row-pairs x 16 cols
                    int q  = tt & 15, ph = tt >> 4;
                    int p0 = ((ph >> l) << (l + 1)) | (ph & ((1 << l) - 1));
                    int p1 = p0 | (1 << l);
                    f4 T = *(const f4*)(twl + (k1 * 8 + ph) * 4); // [p][q] row-major
                    float m0 = mymat[p0 * YSTR + q];
                    float m1 = mymat[p1 * YSTR + q];
                    mymat[p0 * YSTR + q] = T.x * m0 + T.y * m1;
                    mymat[p1 * YSTR + q] = T.z * m0 + T.w * m1;
                }
            }
            // GEMM: D[16 x 16b] = MAT(16x16) x Y[k1 block], K=16 via 4 WMMAs
            v8f c = {};
            const int rowbase = k1 * 16;
#pragma unroll
            for (int kk = 0; kk < 4; ++kk) {
                int K0 = kk * 4 + 2 * kh;
                v2f a, bb;
                a.x  = mymat[mlane * YSTR + K0];
                a.y  = mymat[mlane * YSTR + K0 + 1];
                bb.x = Y[(rowbase + K0) * YSTR + mlane];
                bb.y = Y[(rowbase + K0 + 1) * YSTR + mlane];
                c = __builtin_amdgcn_wmma_f32_16x16x4_f32(
                        false, a, false, bb, (short)0, c, false, false);
            }
#pragma unroll
            for (int j = 0; j < 8; ++j)
                Y[(rowbase + j + 8 * kh) * YSTR + mlane] = c[j];
        }
        __syncthreads();

        // ---------------- Phase B: stages 4..7 (mix n[7:4]) ----------------
        // One 16x16 matrix per c = (d=n[9:8])*16 + (u=n[3:0]).
        for (int blk = 0; blk < 8; ++blk) {
            int cidx = blk * 8 + wv;
            int d = cidx >> 4, u = cidx & 15;
            for (int e = lane; e < 256; e += 32) {
                int p = e >> 4, q = e & 15;
                mymat[p * YSTR + q] = (p == q) ? 1.f : 0.f;
            }
            for (int lp = 0; lp < 4; ++lp) {          // stage l = 4+lp
                const float* twl = twio + (4 + lp) * 2048;
                for (int tt = lane; tt < 128; tt += 32) {
                    int q  = tt & 15, ph = tt >> 4;
                    int p0 = ((ph >> lp) << (lp + 1)) | (ph & ((1 << lp) - 1));
                    int p1 = p0 | (1 << lp);
                    f4 T = *(const f4*)(twl + (d * 128 + ph * 16 + u) * 4);
                    float m0 = mymat[p0 * YSTR + q];
                    float m1 = mymat[p1 * YSTR + q];
                    mymat[p0 * YSTR + q] = T.x * m0 + T.y * m1;
                    mymat[p1 * YSTR + q] = T.z * m0 + T.w * m1;
                }
            }
            v8f c = {};
#pragma unroll
            for (int kk = 0; kk < 4; ++kk) {
                int K0 = kk * 4 + 2 * kh;
                v2f a, bb;
                a.x = mymat[mlane * YSTR + K0];
                a.y = mymat[mlane * YSTR + K0 + 1];
                int r0 = d * 256 + K0 * 16 + u;     // n = d*256 + v*16 + u
                bb.x = Y[r0 * YSTR + mlane];
                bb.y = Y[(r0 + 16) * YSTR + mlane];
                c = __builtin_amdgcn_wmma_f32_16x16x4_f32(
                        false, a, false, bb, (short)0, c, false, false);
            }
#pragma unroll
            for (int j = 0; j < 8; ++j)
                Y[(d * 256 + (j + 8 * kh) * 16 + u) * YSTR + mlane] = c[j];
        }
        __syncthreads();

        // -------- Phase C: stages 8,9 (mix n[9:8]) + ic accumulation --------
        {
            int r = t;                               // r = n[7:0], one per thread
            const float* tw8 = twio + 8 * 2048;
            const float* tw9 = twio + 9 * 2048;
            f4 T8a = *(const f4*)(tw8 + r * 4);          // n9=0 pair
            f4 T8b = *(const f4*)(tw8 + (256 + r) * 4);  // n9=1 pair
            f4 T9a = *(const f4*)(tw9 + r * 4);          // n8=0 pair
            f4 T9b = *(const f4*)(tw9 + (256 + r) * 4);  // n8=1 pair
            for (int b = 0; b < 16; ++b) {
                float v00 = Y[(r      ) * YSTR + b];   // (n9,n8)=(0,0)
                float v01 = Y[(256 + r) * YSTR + b];   // (0,1)
                float v10 = Y[(512 + r) * YSTR + b];   // (1,0)
                float v11 = Y[(768 + r) * YSTR + b];   // (1,1)
                // stage 8: pairs over n8, fixed n9
                float a00 = T8a.x * v00 + T8a.y * v01;
                float a01 = T8a.z * v00 + T8a.w * v01;
                float a10 = T8b.x * v10 + T8b.y * v11;
                float a11 = T8b.z * v10 + T8b.w * v11;
                // stage 9: pairs over n9, fixed n8
                float f00 = T9a.x * a00 + T9a.y * a10;
                float f10 = T9a.z * a00 + T9a.w * a10;
                float f01 = T9b.x * a01 + T9b.y * a11;
                float f11 = T9b.z * a01 + T9b.w * a11;
                ACC[(r      ) * YSTR + b] += f00;
                ACC[(256 + r) * YSTR + b] += f01;
                ACC[(512 + r) * YSTR + b] += f10;
                ACC[(768 + r) * YSTR + b] += f11;
            }
        }
    }
    __syncthreads();

    // epilogue: mean over ic (x 1/16) + bias; out flat layout is [oc][b][n]
    for (int idx = t; idx < 16384; idx += 256) {
        int n = idx & 1023;
        int b = idx >> 10;
        out[(size_t)o * 65536 + (b0 + b) * 1024 + n] =
            ACC[n * YSTR + b] * 0.0625f + bias[o * 1024 + n];
    }
}

extern "C" void kernel_launch(void* const* d_in, const int* in_sizes, int n_in,
                              void* d_out, int out_size, void* d_ws, size_t ws_size,
                              hipStream_t stream) {
    (void)in_sizes; (void)n_in; (void)out_size; (void)d_ws; (void)ws_size;
    const float* x    = (const float*)d_in[0];
    const float* tw   = (const float*)d_in[1];
    const float* bias = (const float*)d_in[2];
    float* out        = (float*)d_out;

    const size_t shmem = (size_t)SHM_FLOATS * sizeof(float);  // 147968 B
    hipFuncSetAttribute((const void*)bconv_butterfly_kernel,
                        hipFuncAttributeMaxDynamicSharedMemorySize, (int)shmem);

    dim3 grid(4, 32, 1);   // 4 batch tiles x 32 output channels
    dim3 block(256, 1, 1); // 8 wave32s
    bconv_butterfly_kernel<<<grid, block, shmem, stream>>>(x, tw, bias, out);
}